// GATLayer_5471788335689
// MI455X (gfx1250) — compile-verified
//
#include <hip/hip_runtime.h>
#include <hip/hip_bf16.h>

typedef float v2f __attribute__((ext_vector_type(2)));
typedef float v8f __attribute__((ext_vector_type(8)));
typedef unsigned int u32x4 __attribute__((ext_vector_type(4)));
typedef int i32x4 __attribute__((ext_vector_type(4)));
typedef int i32x8 __attribute__((ext_vector_type(8)));

#define IN_DIM 128
#define OUT_DIM 64
#define NEG_SLOPE 0.01f

// ---------------------------------------------------------------- init
__global__ void gat_init_kernel(float* __restrict__ out, float* __restrict__ m,
                                float* __restrict__ denom, int n64, int n) {
    int i = blockIdx.x * blockDim.x + threadIdx.x;
    if (i < n64) out[i] = 0.0f;
    if (i < n) {
        m[i] = -__builtin_inff();
        denom[i] = 0.0f;
    }
}

// ---------------------------------------------------------------- GEMM z = h @ W + b
// Block: 256 threads = 8 waves. Tile: 32 rows x 64 cols.
// h tile (32x128 fp32 = 16KB) staged into LDS via the Tensor Data Mover
// (TENSOR_LOAD_TO_LDS, tracked by TENSORcnt), then 8 waves each compute one
// 16x16 D tile with a K=4-step V_WMMA_F32_16X16X4_F32 loop.
__global__ __launch_bounds__(256) void gat_gemm_kernel(
    const float* __restrict__ h, const float* __restrict__ W,
    const float* __restrict__ bias, float* __restrict__ z, int n) {
    __shared__ float hTile[32 * IN_DIM];  // 16 KB

    const int tid = threadIdx.x;
    const long rowBase = (long)blockIdx.x * 32;
    const long elemBase = rowBase * IN_DIM;  // tile is contiguous in h

    if (tid < 32) {  // wave 0 issues the TDM DMA (TDM ignores EXEC within the wave)
        const unsigned lds_off = (unsigned)(size_t)(&hTile[0]);
        const unsigned long long gaddr =
            (unsigned long long)(const void*)(h + elemBase);
        const unsigned rowsAvail = (unsigned)((long)n - rowBase);  // OOB rows -> zero-fill

        // D# group 0: count=1 | lds_addr | global_addr[56:0] | type=2
        u32x4 g0;
        g0.x = 1u;
        g0.y = lds_off;
        g0.z = (unsigned)(gaddr & 0xFFFFFFFFull);
        g0.w = (unsigned)((gaddr >> 32) & 0x01FFFFFFull) | (2u << 30);

        // D# group 1: data_size=2 (4B) | tensor_dim0=128 | tensor_dim1=rowsAvail |
        //             tile_dim0=128 | tile_dim1=32 | tensor_dim0_stride=128
        i32x8 g1;
        g1[0] = (int)(2u << 16);                              // data_size=4B
        g1[1] = (int)((128u & 0xFFFFu) << 16);                // tensor_dim0[15:0]
        g1[2] = (int)(((128u >> 16) & 0xFFFFu) |
                      ((rowsAvail & 0xFFFFu) << 16));         // tdim0[31:16] | tdim1[15:0]
        g1[3] = (int)(((rowsAvail >> 16) & 0xFFFFu) |
                      (128u << 16));                          // tdim1[31:16] | tile_dim0=128
        g1[4] = (int)32u;                                     // tile_dim1=32, tile_dim2=0
        g1[5] = (int)128;                                     // tensor_dim0_stride[31:0]
        g1[6] = 0;                                            // stride[47:32] | dim1_stride lo
        g1[7] = 0;

        i32x4 gz4 = {0, 0, 0, 0};            // groups 2/3 unused (2D tensor)
        i32x8 gz8 = {0, 0, 0, 0, 0, 0, 0, 0};  // extra operand of 6-arg builtin

        __builtin_amdgcn_tensor_load_to_lds(g0, g1, gz4, gz4, gz8, 0);
        __builtin_amdgcn_s_wait_tensorcnt(0);
    }
    __syncthreads();

    const int wave = tid >> 5;
    const int lane = tid & 31;
    const int rg = wave >> 2;         // 0..1
    const int cg = wave & 3;          // 0..3
    const int mrow = lane & 15;       // M (A) / N (B,D) index within tile
    const int kk = (lane >> 4) << 1;  // lanes 16-31 hold K=2,3
    const int col = cg * 16 + mrow;

    v8f c = {0.f, 0.f, 0.f, 0.f, 0.f, 0.f, 0.f, 0.f};
    const float* hrow = hTile + (rg * 16 + mrow) * IN_DIM;

    for (int k = 0; k < IN_DIM; k += 4) {
        v2f a;
        a.x = hrow[k + kk];
        a.y = hrow[k + kk + 1];
        v2f bfr;
        bfr.x = W[(k + kk) * OUT_DIM + col];
        bfr.y = W[(k + kk + 1) * OUT_DIM + col];
        c = __builtin_amdgcn_wmma_f32_16x16x4_f32(
            /*neg_a=*/false, a, /*neg_b=*/false, bfr,
            /*c_mod=*/(short)0, c, /*reuse_a=*/false, /*reuse_b=*/false);
    }

    const float bv = bias[col];
    const long row0 = rowBase + rg * 16 + (lane >> 4) * 8;  // D: VGPR i -> M = i + 8*(lane/16)
    for (int i = 0; i < 8; ++i) {
        long r = row0 + i;
        if (r < n) z[r * OUT_DIM + col] = c[i] + bv;
    }
}

// ---------------------------------------------------------------- per-node scores
__global__ void gat_score_kernel(const float* __restrict__ z,
                                 const float* __restrict__ a_w,
                                 float* __restrict__ s_src,
                                 float* __restrict__ s_dst, int n) {
    int i = blockIdx.x * blockDim.x + threadIdx.x;
    if (i >= n) return;
    const float* zr = z + (long)i * OUT_DIM;
    float as = 0.f, ad = 0.f;
#pragma unroll 8
    for (int j = 0; j < OUT_DIM; ++j) {
        float v = zr[j];
        as += v * a_w[j];
        ad += v * a_w[OUT_DIM + j];
    }
    s_src[i] = as;
    s_dst[i] = ad;
}

// ---------------------------------------------------------------- float atomic max
__device__ inline void atomicMaxFloat(float* addr, float val) {
    if (val >= 0.0f)
        atomicMax((int*)addr, __float_as_int(val));
    else
        atomicMin((unsigned int*)addr, (unsigned int)__float_as_int(val));
}

// ---------------------------------------------------------------- edge score + segment max
__global__ void gat_edge_score_kernel(const int* __restrict__ src,
                                      const int* __restrict__ dst,
                                      const float* __restrict__ s_src,
                                      const float* __restrict__ s_dst,
                                      const float* __restrict__ a_b,
                                      float* __restrict__ ebuf,
                                      float* __restrict__ m, int ne) {
    int e = blockIdx.x * blockDim.x + threadIdx.x;
    if (e >= ne) return;
    int s = src[e], d = dst[e];
    float v = s_src[s] + s_dst[d] + a_b[0];
    v = (v > 0.0f) ? v : NEG_SLOPE * v;
    ebuf[e] = v;
    atomicMaxFloat(m + d, v);
}

// ---------------------------------------------------------------- exp + segment sum
__global__ void gat_edge_exp_kernel(const int* __restrict__ dst,
                                    const float* __restrict__ m,
                                    float* __restrict__ ebuf,
                                    float* __restrict__ denom, int ne) {
    int e = blockIdx.x * blockDim.x + threadIdx.x;
    if (e >= ne) return;
    int d = dst[e];
    float md = m[d];
    if (!__builtin_isfinite(md)) md = 0.0f;
    float ex = __expf(ebuf[e] - md);
    ebuf[e] = ex;
    atomicAdd(denom + d, ex);
}

// ---------------------------------------------------------------- message scatter
// One wave (32 lanes) per edge; each lane handles 2 consecutive output cols.
__global__ __launch_bounds__(256) void gat_scatter_kernel(
    const int* __restrict__ src, const int* __restrict__ dst,
    const float* __restrict__ z, const float* __restrict__ ebuf,
    const float* __restrict__ denom, float* __restrict__ out, int ne) {
    int t = blockIdx.x * 256 + threadIdx.x;
    int e = t >> 5;
    int lane = t & 31;
    if (e >= ne) return;
    int s = src[e], d = dst[e];
    float den = denom[d];
    float alpha = ebuf[e] / ((den > 0.0f) ? den : 1.0f);
    float2 v = ((const float2*)(z + (long)s * OUT_DIM))[lane];
    float* o = out + (long)d * OUT_DIM + lane * 2;
    atomicAdd(o, alpha * v.x);
    atomicAdd(o + 1, alpha * v.y);
}

// ---------------------------------------------------------------- launcher
extern "C" void kernel_launch(void* const* d_in, const int* in_sizes, int n_in,
                              void* d_out, int out_size, void* d_ws, size_t ws_size,
                              hipStream_t stream) {
    const float* h    = (const float*)d_in[0];
    const int*   src  = (const int*)  d_in[1];
    const int*   dst  = (const int*)  d_in[2];
    const float* W    = (const float*)d_in[3];
    const float* bias = (const float*)d_in[4];
    const float* a_w  = (const float*)d_in[5];
    const float* a_b  = (const float*)d_in[6];
    float* out = (float*)d_out;

    const int n  = in_sizes[0] / IN_DIM;  // 100000
    const int ne = in_sizes[1];           // 1600000

    // Workspace layout (floats): z[n*64] | s_src[n] | s_dst[n] | m[n] | denom[n] | ebuf[ne]
    float* ws    = (float*)d_ws;
    float* z     = ws;
    float* s_src = z + (size_t)n * OUT_DIM;
    float* s_dst = s_src + n;
    float* mbuf  = s_dst + n;
    float* denom = mbuf + n;
    float* ebuf  = denom + n;

    const int n64 = n * OUT_DIM;

    gat_init_kernel<<<(n64 + 255) / 256, 256, 0, stream>>>(out, mbuf, denom, n64, n);

    gat_gemm_kernel<<<(n + 31) / 32, 256, 0, stream>>>(h, W, bias, z, n);

    gat_score_kernel<<<(n + 255) / 256, 256, 0, stream>>>(z, a_w, s_src, s_dst, n);

    gat_edge_score_kernel<<<(ne + 255) / 256, 256, 0, stream>>>(
        src, dst, s_src, s_dst, a_b, ebuf, mbuf, ne);

    gat_edge_exp_kernel<<<(ne + 255) / 256, 256, 0, stream>>>(dst, mbuf, ebuf, denom, ne);

    long scatterThreads = (long)ne * 32;
    gat_scatter_kernel<<<(unsigned)((scatterThreads + 255) / 256), 256, 0, stream>>>(
        src, dst, z, ebuf, denom, out, ne);
}